// ROIAlign_22454089023570
// MI455X (gfx1250) — compile-verified
//
#include <hip/hip_runtime.h>
#include <stdint.h>

// ---------------------------------------------------------------------------
// ROI Align for MI455X (gfx1250): TDM-staged LDS sampling.
//   - tensor_load_to_lds DMAs a 3D tile (x-span x y-rows x 8 channels) of the
//     ROI's bounding patch into LDS (double buffered, TENSORcnt pipelined).
//   - Bilinear taps are then LDS reads instead of 103M scattered global loads.
// ---------------------------------------------------------------------------

typedef __attribute__((ext_vector_type(4))) unsigned int v4u;
typedef __attribute__((ext_vector_type(8))) int          v8i;
typedef __attribute__((ext_vector_type(4))) int          v4i;

#define C_TOT   256
#define H_IN    200
#define W_IN    200
#define PLANE   (H_IN * W_IN)      // 40000
#define OUTD    7
#define NP      49                  // 7x7 outputs
#define NS      14                  // samples per axis (7 bins x 2 subsamples)
#define CH      8                   // channels per TDM stage
#define CSPLIT  4                   // channel-split blocks per ROI
#define CPB     (C_TOT / CSPLIT)    // 64 channels per block
#define NSTAGE  (CPB / CH)          // 8 stages per block
#define SPAN_MAX 48                 // max patch extent (192*0.25 + slack)
#define PATCH_FLOATS (CH * SPAN_MAX * SPAN_MAX)   // 18432 floats = 72KB

struct RoiMeta {
    int   xl[NS], xh[NS], yl[NS], yh[NS];   // tap coords relative to patch
    float lx[NS], ly[NS];                   // fractional weights
    int   xlo, ylo, spanX, spanY, b;
    float valid;
};

__device__ __forceinline__ unsigned rfl(unsigned x) {
    return (unsigned)__builtin_amdgcn_readfirstlane((int)x);
}

// Build the Tensor DMA Descriptor (D#) per CDNA5 ISA ch.8 and issue the load.
// 3D tile: dim0 = x (stride 200), dim1 = y (stride 40000), dim2 = channel
// (uses tensor_dim1_stride = PLANE for the 3rd loop per the address formula).
__device__ __forceinline__ void tdm_issue(unsigned lds_addr, uint64_t gaddr,
                                          unsigned spanX, unsigned spanY) {
    unsigned alo = rfl((unsigned)gaddr);
    unsigned ahi = rfl((unsigned)(gaddr >> 32));
    unsigned sx  = rfl(spanX);
    unsigned sy  = rfl(spanY);
    unsigned la  = rfl(lds_addr);

    v4u g0;
    g0.x = 1u;                                    // count=1, user descriptor
    g0.y = la;                                    // lds_addr (bytes)
    g0.z = alo;                                   // global_addr[31:0]
    g0.w = (ahi & 0x01FFFFFFu) | 0x80000000u;     // global_addr[56:32] | type=2

    v8i g1;
    g1.s0 = (int)(2u << 16);                                    // data_size=4B
    g1.s1 = (int)(((unsigned)W_IN & 0xFFFFu) << 16);            // tensor_dim0 lo16
    g1.s2 = (int)((((unsigned)W_IN >> 16) & 0xFFFFu) |
                  (((unsigned)H_IN & 0xFFFFu) << 16));          // td0 hi | td1 lo
    g1.s3 = (int)((((unsigned)H_IN >> 16) & 0xFFFFu) |
                  (sx << 16));                                  // td1 hi | tile_dim0
    g1.s4 = (int)(sy | ((unsigned)CH << 16));                   // tile_dim1 | tile_dim2
    g1.s5 = (int)(unsigned)W_IN;                                // dim0_stride lo32
    g1.s6 = (int)((((unsigned)PLANE) & 0xFFFFu) << 16);         // dim0_str hi=0 | dim1_str lo16
    g1.s7 = (int)(((unsigned)PLANE) >> 16);                     // dim1_stride[47:16]

    v4i g2;
    g2.x = C_TOT;   // tensor_dim2 (channel extent, never exceeded)
    g2.y = 0;       // tensor_dim3 (unused)
    g2.z = 0;       // tensor_dim2_stride lo (4th dim unused)
    g2.w = 0;       // tile_dim3 = 0 (unused)

    v4i g3;
    g3.x = 0; g3.y = 0; g3.z = 0; g3.w = 0;       // dims 3/4 unused

    v8i g4;         // extra descriptor group required by the 6-arg builtin
    g4.s0 = 0; g4.s1 = 0; g4.s2 = 0; g4.s3 = 0;   // (unused: zero-filled)
    g4.s4 = 0; g4.s5 = 0; g4.s6 = 0; g4.s7 = 0;

    __builtin_amdgcn_tensor_load_to_lds(g0, g1, g2, g3, g4, 0);
}

__global__ __launch_bounds__(256) void roi_align_tdm(
    const float* __restrict__ inp, const float* __restrict__ rois,
    float* __restrict__ out) {
    __shared__ __align__(16) float patch[2][PATCH_FLOATS];
    __shared__ RoiMeta m;

    const int k     = blockIdx.x;
    const int cbase = blockIdx.y * CPB;

    // ---- Per-ROI setup (thread 0, ~100 scalar ops) ----
    if (threadIdx.x == 0) {
        const float* r = rois + 5 * k;
        int   b  = (int)r[0];
        float x1 = r[1] * 0.25f, y1 = r[2] * 0.25f;
        float x2 = r[3] * 0.25f, y2 = r[4] * 0.25f;
        float rw = fmaxf(x2 - x1, 1.0f), rh = fmaxf(y2 - y1, 1.0f);
        float bw = rw / (float)OUTD, bh = rh / (float)OUTD;
        bool valid = true;
        int xi_[NS], xh_[NS], yi_[NS], yh_[NS];
        for (int s = 0; s < NS; ++s) {
            float fp = (float)(s >> 1);
            float fi = (float)(s & 1) + 0.5f;
            float xs = x1 + bw * fp + (bw * 0.5f) * fi;
            float ys = y1 + bh * fp + (bh * 0.5f) * fi;
            valid = valid && (xs >= -1.0f) && (xs <= (float)W_IN)
                          && (ys >= -1.0f) && (ys <= (float)H_IN);
            float xc = fminf(fmaxf(xs, 0.0f), (float)(W_IN - 1));
            float yc = fminf(fmaxf(ys, 0.0f), (float)(H_IN - 1));
            float xf = floorf(xc), yf = floorf(yc);
            m.lx[s] = xc - xf;
            m.ly[s] = yc - yf;
            xi_[s] = (int)xf;  xh_[s] = min((int)xf + 1, W_IN - 1);
            yi_[s] = (int)yf;  yh_[s] = min((int)yf + 1, H_IN - 1);
        }
        int xlo = xi_[0], ylo = yi_[0];                 // coords are monotone
        m.xlo   = xlo;  m.ylo = ylo;
        m.spanX = min(xh_[NS - 1] - xlo + 1, SPAN_MAX);
        m.spanY = min(yh_[NS - 1] - ylo + 1, SPAN_MAX);
        for (int s = 0; s < NS; ++s) {
            m.xl[s] = xi_[s] - xlo;  m.xh[s] = xh_[s] - xlo;
            m.yl[s] = yi_[s] - ylo;  m.yh[s] = yh_[s] - ylo;
        }
        m.b = b;
        m.valid = valid ? 1.0f : 0.0f;
    }
    __syncthreads();

    const int   spanX = m.spanX, spanY = m.spanY;
    const float valid = m.valid;
    const uint64_t base = (uint64_t)(uintptr_t)inp +
        4ull * ((uint64_t)(unsigned)(m.b * C_TOT + cbase) * (uint64_t)PLANE +
                (uint64_t)m.ylo * (uint64_t)W_IN + (uint64_t)m.xlo);
    const uint64_t stage_step = (uint64_t)CH * (uint64_t)PLANE * 4ull;
    const bool issuer = (threadIdx.x < 32);   // wave 0 owns TENSORcnt

    if (issuer) {
        tdm_issue((unsigned)(uintptr_t)&patch[0][0], base,
                  (unsigned)spanX, (unsigned)spanY);
    }

    for (int s = 0; s < NSTAGE; ++s) {
        if (issuer) {
            if (s + 1 < NSTAGE) {
                // Prefetch next stage into the other buffer, then wait for the
                // current stage (TENSORcnt is in-order within a wave).
                tdm_issue((unsigned)(uintptr_t)&patch[(s + 1) & 1][0],
                          base + (uint64_t)(s + 1) * stage_step,
                          (unsigned)spanX, (unsigned)spanY);
                __builtin_amdgcn_s_wait_tensorcnt(1);
            } else {
                __builtin_amdgcn_s_wait_tensorcnt(0);
            }
        }
        __syncthreads();   // stage s resident in LDS for all waves

        const float* P = patch[s & 1];
        for (int t = threadIdx.x; t < CH * NP; t += 256) {
            int c  = t / NP;
            int p  = t - c * NP;
            int py = p / OUTD;
            int px = p - py * OUTD;
            const float* Pc = P + c * spanY * spanX;
            float acc = 0.0f;
#pragma unroll
            for (int i = 0; i < 2; ++i) {
                int   ysi = 2 * py + i;
                float lyv = m.ly[ysi], hyv = 1.0f - lyv;
                const float* r0 = Pc + m.yl[ysi] * spanX;
                const float* r1 = Pc + m.yh[ysi] * spanX;
#pragma unroll
                for (int j = 0; j < 2; ++j) {
                    int   xsi = 2 * px + j;
                    float lxv = m.lx[xsi], hxv = 1.0f - lxv;
                    int   xl = m.xl[xsi], xh = m.xh[xsi];
                    acc += hyv * (hxv * r0[xl] + lxv * r0[xh]) +
                           lyv * (hxv * r1[xl] + lxv * r1[xh]);
                }
            }
            out[(size_t)(k * C_TOT + cbase + s * CH + c) * NP + p] =
                acc * 0.25f * valid;
        }
        __syncthreads();   // buffer (s&1) free for stage s+2's TDM write
    }
}

extern "C" void kernel_launch(void* const* d_in, const int* in_sizes, int n_in,
                              void* d_out, int out_size, void* d_ws, size_t ws_size,
                              hipStream_t stream) {
    (void)n_in; (void)out_size; (void)d_ws; (void)ws_size;
    const float* inp  = (const float*)d_in[0];
    const float* rois = (const float*)d_in[1];
    float* out = (float*)d_out;
    int K = in_sizes[1] / 5;   // 512
    dim3 grid((unsigned)K, CSPLIT, 1);
    roi_align_tdm<<<grid, 256, 0, stream>>>(inp, rois, out);
}